// DecisionTreePolicy_8727373546182
// MI455X (gfx1250) — compile-verified
//
#include <hip/hip_runtime.h>
#include <hip/hip_bf16.h>
#include <stdint.h>

#define NFEAT 256
#define NACT  64
#define LDS_NODES 4095        // complete-tree levels 0..11; 4 arrays -> 64 KB LDS
#define LDS_PAD   4096
#define LDS_LEVELS 12         // steps guaranteed resolvable from LDS (complete tree)
#define ROWS_PER_THREAD 4     // ILP across independent traversal chains
#define BLOCK 256

typedef __attribute__((ext_vector_type(4))) unsigned int u32x4_t;
typedef __attribute__((ext_vector_type(8))) int          i32x8_t;
typedef __attribute__((ext_vector_type(4))) int          i32x4_t;
typedef __attribute__((ext_vector_type(4))) float        f32x4_t;  // native clang vector

#if defined(__gfx1250__) && __has_builtin(__builtin_amdgcn_tensor_load_to_lds)
#define HAVE_TDM 1
#else
#define HAVE_TDM 0
#endif

// 1-D DMA of `ndw` DWORDs from global memory to LDS via the Tensor Data Mover.
// D# layout per CDNA5 ISA (groups 2/3 zero => <=2-D tensor).
__device__ __forceinline__ void tdm_load_dwords_to_lds(uint32_t lds_byte_off,
                                                       const void* gsrc,
                                                       uint32_t ndw) {
#if HAVE_TDM
  uint64_t ga = (uint64_t)(uintptr_t)gsrc;
  u32x4_t g0;
  g0[0] = 1u;                                   // count=1 (valid descriptor), user mode
  g0[1] = lds_byte_off;                         // lds_addr (bytes)
  g0[2] = (uint32_t)(ga & 0xFFFFFFFFu);         // global_addr[31:0]
  g0[3] = (uint32_t)((ga >> 32) & 0x01FFFFFFu)  // global_addr[56:32]
        | (2u << 30);                           // type = 2 ("image")

  i32x8_t g1;
  g1[0] = (int)(2u << 16);                      // wg_mask=0, data_size=2 (4 B), no pad
  g1[1] = (int)((ndw & 0xFFFFu) << 16);         // tensor_dim0[15:0]  @ bits 63:48
  g1[2] = (int)(((ndw >> 16) & 0xFFFFu)         // tensor_dim0[31:16]
        | (1u << 16));                          // tensor_dim1 = 1
  g1[3] = (int)((ndw & 0xFFFFu) << 16);         // tensor_dim1[31:16]=0, tile_dim0=ndw
  g1[4] = 0;                                    // tile_dim1=0, tile_dim2=0 (1-D tile)
  g1[5] = (int)ndw;                             // tensor_dim0_stride (don't-care for 1-D)
  g1[6] = 0;
  g1[7] = 0;

  i32x4_t gz4 = {0, 0, 0, 0};
  i32x8_t gz8 = {0, 0, 0, 0, 0, 0, 0, 0};
  // 6-arg form (upstream clang-23 / therock HIP headers).
  __builtin_amdgcn_tensor_load_to_lds(g0, g1, gz4, gz4, gz8, 0);
#else
  (void)lds_byte_off; (void)gsrc; (void)ndw;
#endif
}

__device__ __forceinline__ void wait_tensorcnt0() {
#if defined(__gfx1250__)
#if __has_builtin(__builtin_amdgcn_s_wait_tensorcnt)
  __builtin_amdgcn_s_wait_tensorcnt(0);
#else
  asm volatile("s_wait_tensorcnt 0" ::: "memory");
#endif
#endif
}

__global__ __launch_bounds__(BLOCK) void DecisionTreePolicy_kernel(
    const float* __restrict__ obs,          // [B, 256]
    const int*   __restrict__ features,     // [32767]
    const float* __restrict__ thresholds,   // [32767]
    const int*   __restrict__ cleft,        // [32767]
    const int*   __restrict__ cright,       // [32767]
    const float* __restrict__ leaf_logits,  // [32767, 64]
    float*       __restrict__ out,          // [B, 64]
    int batch)
{
  __shared__ int   s_feat[LDS_PAD];
  __shared__ float s_thr [LDS_PAD];
  __shared__ int   s_cl  [LDS_PAD];
  __shared__ int   s_cr  [LDS_PAD];

  // The TDM writes the shared arrays behind the compiler's back. Without a
  // visible store LLVM proves them store-free and folds the LDS reads away.
  // This never-taken, runtime-opaque store keeps them "written".
  if (batch < 0) {
    s_feat[threadIdx.x] = batch;
    s_thr [threadIdx.x] = (float)batch;
    s_cl  [threadIdx.x] = batch;
    s_cr  [threadIdx.x] = batch;
  }

#if HAVE_TDM
  if (threadIdx.x == 0) {
    tdm_load_dwords_to_lds((uint32_t)(uintptr_t)(void*)s_feat, features,   LDS_NODES);
    tdm_load_dwords_to_lds((uint32_t)(uintptr_t)(void*)s_thr,  thresholds, LDS_NODES);
    tdm_load_dwords_to_lds((uint32_t)(uintptr_t)(void*)s_cl,   cleft,      LDS_NODES);
    tdm_load_dwords_to_lds((uint32_t)(uintptr_t)(void*)s_cr,   cright,     LDS_NODES);
  }
  // Every wave waits (no-op for waves with TENSORcnt==0; covers the case where
  // the compiler predicates instead of branching, since TDM ignores EXEC).
  wait_tensorcnt0();
#else
  for (int i = (int)threadIdx.x; i < LDS_NODES; i += BLOCK) {
    s_feat[i] = features[i];
    s_thr[i]  = thresholds[i];
    s_cl[i]   = cleft[i];
    s_cr[i]   = cright[i];
  }
#endif
  // Opaque barrier: nothing may be folded/hoisted across the DMA completion.
  asm volatile("" ::: "memory");
  __syncthreads();

  const int tid    = (int)(blockIdx.x * blockDim.x + threadIdx.x);
  const int stride = (int)(gridDim.x * blockDim.x);

  int          node[ROWS_PER_THREAD];
  bool         act [ROWS_PER_THREAD];
  const float* orow[ROWS_PER_THREAD];
  int          row [ROWS_PER_THREAD];

#pragma unroll
  for (int j = 0; j < ROWS_PER_THREAD; ++j) {
    row[j]  = tid + j * stride;
    bool ok = row[j] < batch;
    orow[j] = obs + (size_t)(ok ? row[j] : 0) * NFEAT;
    node[j] = 0;
    act[j]  = ok;
    if (ok) {
      __builtin_prefetch(orow[j], 0, 0);                 // global_prefetch_b8
      __builtin_prefetch(orow[j] + NFEAT / 2, 0, 0);
    }
  }

  // ---- Phase 1: node data purely from LDS (levels 0..11 of a complete tree).
  // The body touches only addrspace(3) arrays, so nothing can be merged into a
  // flat pointer-select: these lower to ds_load_b32.
#pragma unroll 1
  for (int level = 0; level < LDS_LEVELS; ++level) {
#pragma unroll
    for (int j = 0; j < ROWS_PER_THREAD; ++j) {
      const int n = node[j];
      if (act[j] && n < LDS_NODES) {
        const int   l = s_cl[n];
        const int   r = s_cr[n];
        const int   f = s_feat[n];
        const float t = s_thr[n];
        if (l != r) {
          const float o = orow[j][f];
          node[j] = (o <= t) ? l : r;
        } else {
          act[j] = false;       // leaf reached inside the LDS range
        }
      }
    }
  }

  // ---- Phase 2: remaining levels purely from the L2-resident global tables
  // (valid for any node id, so this is correct even for non-complete trees).
#pragma unroll 1
  for (int level = 0; level < 28; ++level) {
    bool any = false;
#pragma unroll
    for (int j = 0; j < ROWS_PER_THREAD; ++j) {
      if (act[j]) {
        const int   n = node[j];
        const int   l = cleft[n];
        const int   r = cright[n];
        const int   f = features[n];
        const float t = thresholds[n];
        if (l != r) {
          const float o = orow[j][f];
          node[j] = (o <= t) ? l : r;
        } else {
          act[j] = false;       // node[j] is the leaf id
        }
      }
      any |= act[j];
    }
    if (!any) break;
  }

  // Gather the 64-float logits row per observation (L2-resident table).
  // Output is write-once streaming data: use non-temporal stores (TH=NT) so the
  // 64 MB result doesn't evict the hot leaf_logits/obs lines from L2.
#pragma unroll
  for (int j = 0; j < ROWS_PER_THREAD; ++j) {
    if (row[j] < batch) {
      const f32x4_t* __restrict__ src =
          (const f32x4_t*)(leaf_logits + (size_t)node[j] * NACT);
      f32x4_t* __restrict__ dst = (f32x4_t*)(out + (size_t)row[j] * NACT);
#pragma unroll
      for (int k = 0; k < NACT / 4; ++k) {
        __builtin_nontemporal_store(src[k], &dst[k]);
      }
    }
  }
}

extern "C" void kernel_launch(void* const* d_in, const int* in_sizes, int n_in,
                              void* d_out, int out_size, void* d_ws, size_t ws_size,
                              hipStream_t stream) {
  const float* obs         = (const float*)d_in[0];
  const int*   features    = (const int*)  d_in[1];
  const float* thresholds  = (const float*)d_in[2];
  const int*   cleft       = (const int*)  d_in[3];
  const int*   cright      = (const int*)  d_in[4];
  const float* leaf_logits = (const float*)d_in[5];
  float*       out         = (float*)d_out;

  const int batch = in_sizes[0] / NFEAT;
  const int total_threads = (batch + ROWS_PER_THREAD - 1) / ROWS_PER_THREAD;
  const int grid = (total_threads + BLOCK - 1) / BLOCK;

  DecisionTreePolicy_kernel<<<grid, BLOCK, 0, stream>>>(
      obs, features, thresholds, cleft, cright, leaf_logits, out, batch);
}